// OnlineFlashMTPModel_61804579389843
// MI455X (gfx1250) — compile-verified
//
#include <hip/hip_runtime.h>
#include <math.h>

typedef __attribute__((ext_vector_type(2))) float v2f;
typedef __attribute__((ext_vector_type(8))) float v8f;

// Problem shape (fixed by the reference setup)
constexpr int H  = 32;
constexpr int N  = 256;
constexpr int BS = 32;
constexpr int W  = 128;
constexpr int T  = 4096;
constexpr int D  = 128;
constexpr int L  = W + BS;          // 160 keys per tile
constexpr int QROWS = BS;           // 32 query rows per tile

// LDS strides (floats). +4 keeps float4 alignment and avoids bank conflicts
// for column-strided WMMA fragment reads (stride*4 bytes % 64 banks cycles).
constexpr int QS_STR = D + 4;       // 132
constexpr int KV_STR = D + 4;       // 132
constexpr int SS_STR = L + 4;       // 164

constexpr int SMEM_FLOATS = QROWS * QS_STR + L * KV_STR + QROWS * SS_STR;

__device__ __forceinline__ float wave_max32(float v) {
#pragma unroll
  for (int off = 16; off >= 1; off >>= 1)
    v = fmaxf(v, __shfl_xor(v, off, 32));
  return v;
}
__device__ __forceinline__ float wave_sum32(float v) {
#pragma unroll
  for (int off = 16; off >= 1; off >>= 1)
    v += __shfl_xor(v, off, 32);
  return v;
}

__global__ __launch_bounds__(256) void flash_mtp_kernel(
    const float* __restrict__ q,        // (H, N*BS, D)
    const float* __restrict__ k_ctx,    // (H, T, D)
    const float* __restrict__ v_ctx,    // (H, T, D)
    const float* __restrict__ k_draft,  // (H, N*BS, D)
    const float* __restrict__ v_draft,  // (H, N*BS, D)
    const int*   __restrict__ anchors,  // (N)
    const int*   __restrict__ keep,     // (N)
    float* __restrict__ out)            // (H, N*BS, D)
{
  extern __shared__ float smem[];
  float* qs  = smem;                          // [QROWS][QS_STR]
  float* kvs = qs + QROWS * QS_STR;           // [L][KV_STR]  (K, later V)
  float* ss  = kvs + L * KV_STR;              // [QROWS][SS_STR]

  const int hn   = blockIdx.x;
  const int h    = hn / N;
  const int n    = hn % N;
  const int tid  = threadIdx.x;
  const int wave = tid >> 5;
  const int lane = tid & 31;
  const int lane15 = lane & 15;
  const int hi     = lane >> 4;     // 0 for lanes 0-15, 1 for lanes 16-31
  const int khalf  = hi * 2;        // f32 WMMA operand K-pair select

  const int   anchor = anchors[n];
  const float keepf  = keep[n] ? 1.0f : 0.0f;
  const float scale  = 0.08838834764831845f;  // 1/sqrt(128)

  const size_t qrow0 = (size_t)h * (N * BS) + (size_t)n * BS;  // first q/draft row
  const float* qbase  = q       + qrow0 * D;
  const float* kdbase = k_draft + qrow0 * D;
  const float* vdbase = v_draft + qrow0 * D;
  const float* kcbase = k_ctx + (size_t)h * T * D;
  const float* vcbase = v_ctx + (size_t)h * T * D;

  // ---- Stage Q into LDS (32 rows x 128 = 1024 float4) ----
  for (int i = tid; i < QROWS * (D / 4); i += 256) {
    int r = i >> 5, c4 = (i & 31) * 4;
    float4 v = *(const float4*)(qbase + (size_t)r * D + c4);
    *(float4*)(qs + r * QS_STR + c4) = v;
  }
  // ---- Stage gathered K into LDS (160 rows x 128) ----
  for (int i = tid; i < L * (D / 4); i += 256) {
    int r = i >> 5, c4 = (i & 31) * 4;
    const float* src;
    if (r < W) {
      int idx = anchor - W + r;
      idx = idx < 0 ? 0 : (idx > T - 1 ? T - 1 : idx);
      src = kcbase + (size_t)idx * D + c4;
    } else {
      src = kdbase + (size_t)(r - W) * D + c4;
    }
    *(float4*)(kvs + r * KV_STR + c4) = *(const float4*)src;
  }
  __syncthreads();

  // ---- Phase S: S = scale * Q * K^T  (32x160 = 2x10 tiles of 16x16) ----
  for (int t = wave; t < 20; t += 8) {
    const int m0 = (t / 10) * 16;
    const int j0 = (t % 10) * 16;
    const float* qrow = qs  + (m0 + lane15) * QS_STR;   // A: Q row
    const float* krow = kvs + (j0 + lane15) * KV_STR;   // B: K row (transposed access)
    v8f c = {0.f, 0.f, 0.f, 0.f, 0.f, 0.f, 0.f, 0.f};
#pragma unroll 8
    for (int k = 0; k < D; k += 4) {
      v2f a = *(const v2f*)(qrow + k + khalf);
      v2f b = *(const v2f*)(krow + k + khalf);
      c = __builtin_amdgcn_wmma_f32_16x16x4_f32(false, a, false, b,
                                                (short)0, c, false, false);
    }
#pragma unroll
    for (int i = 0; i < 8; ++i)
      ss[(m0 + 8 * hi + i) * SS_STR + j0 + lane15] = c[i] * scale;
  }
  __syncthreads();

  // ---- Softmax with masking: one wave per 4 rows, 32 lanes x 5 cols ----
  {
    const int jmin = W - anchor;  // context col j valid iff j >= jmin (for j < W)
    for (int r = wave * 4; r < wave * 4 + 4; ++r) {
      float vals[5];
      float mx = -1e30f;
#pragma unroll
      for (int it = 0; it < 5; ++it) {
        int j = lane + it * 32;                 // exactly covers 0..159
        bool ok = ((j >= W) || (j >= jmin)) && (keepf != 0.0f);
        float s = ss[r * SS_STR + j];
        vals[it] = ok ? s : -1e30f;
        mx = fmaxf(mx, vals[it]);
      }
      mx = wave_max32(mx);
      float sum = 0.f;
#pragma unroll
      for (int it = 0; it < 5; ++it) {
        vals[it] = expf(vals[it] - mx);
        sum += vals[it];
      }
      sum = wave_sum32(sum);
      float inv = 1.0f / sum;
#pragma unroll
      for (int it = 0; it < 5; ++it)
        ss[r * SS_STR + lane + it * 32] = vals[it] * inv;
    }
  }

  // ---- Stage gathered V into LDS (reuse kvs; safe: S phase done) ----
  for (int i = tid; i < L * (D / 4); i += 256) {
    int r = i >> 5, c4 = (i & 31) * 4;
    const float* src;
    if (r < W) {
      int idx = anchor - W + r;
      idx = idx < 0 ? 0 : (idx > T - 1 ? T - 1 : idx);
      src = vcbase + (size_t)idx * D + c4;
    } else {
      src = vdbase + (size_t)(r - W) * D + c4;
    }
    *(float4*)(kvs + r * KV_STR + c4) = *(const float4*)src;
  }
  __syncthreads();

  // ---- Phase O: O = P * V  (32x128 = 2x8 tiles of 16x16), K-dim = 160 ----
  for (int t = wave; t < 16; t += 8) {
    const int m0 = (t / 8) * 16;
    const int n0 = (t % 8) * 16;
    const float* prow = ss + (m0 + lane15) * SS_STR;    // A: P row
    v8f c = {0.f, 0.f, 0.f, 0.f, 0.f, 0.f, 0.f, 0.f};
#pragma unroll 8
    for (int k = 0; k < L; k += 4) {
      v2f a = *(const v2f*)(prow + k + khalf);
      v2f b;
      b.x = kvs[(k + khalf + 0) * KV_STR + n0 + lane15]; // B: V direct (row = key pos)
      b.y = kvs[(k + khalf + 1) * KV_STR + n0 + lane15];
      c = __builtin_amdgcn_wmma_f32_16x16x4_f32(false, a, false, b,
                                                (short)0, c, false, false);
    }
#pragma unroll
    for (int i = 0; i < 8; ++i) {
      int row = m0 + 8 * hi + i;
      out[(qrow0 + row) * D + n0 + lane15] = c[i] * keepf;
    }
  }
}

extern "C" void kernel_launch(void* const* d_in, const int* in_sizes, int n_in,
                              void* d_out, int out_size, void* d_ws, size_t ws_size,
                              hipStream_t stream) {
  const float* q       = (const float*)d_in[0];
  const float* k_ctx   = (const float*)d_in[1];
  const float* v_ctx   = (const float*)d_in[2];
  const float* k_draft = (const float*)d_in[3];
  const float* v_draft = (const float*)d_in[4];
  const int*   anchors = (const int*)d_in[5];
  const int*   keep    = (const int*)d_in[6];
  float*       out     = (float*)d_out;

  const size_t smem_bytes = (size_t)SMEM_FLOATS * sizeof(float);  // ~120 KB
  dim3 grid(H * N);   // 8192 (h,n) tiles
  dim3 block(256);    // 8 wave32s
  flash_mtp_kernel<<<grid, block, smem_bytes, stream>>>(
      q, k_ctx, v_ctx, k_draft, v_draft, anchors, keep, out);
}